// KarplusStrongResonator_71201967833415
// MI455X (gfx1250) — compile-verified
//
#include <hip/hip_runtime.h>
#include <math.h>

#define T_LEN 8192
#define NROWS 128
#define NDELAY 360   // MAX_DELAY - MIN_DELAY

// ---- gfx1250 async global<->LDS builtins (probe-confirmed in round 1/2) ----
typedef int v4i __attribute__((vector_size(16)));
typedef __attribute__((address_space(1))) v4i* as1_v4i_ptr;
typedef __attribute__((address_space(3))) v4i* as3_v4i_ptr;
typedef __attribute__((address_space(3))) float* as3_float_ptr;
typedef unsigned int u32x4 __attribute__((vector_size(16)));
typedef unsigned int u32x8 __attribute__((vector_size(32)));

#if defined(__AMDGCN__) && \
    __has_builtin(__builtin_amdgcn_global_load_async_to_lds_b128) && \
    __has_builtin(__builtin_amdgcn_global_store_async_from_lds_b128)
#define HAVE_ASYNC_LDS 1
#else
#define HAVE_ASYNC_LDS 0
#endif

#if defined(__AMDGCN__) && __has_builtin(__builtin_amdgcn_s_wait_asynccnt)
#define WAIT_ASYNC0() __builtin_amdgcn_s_wait_asynccnt(0)
#else
#define WAIT_ASYNC0() asm volatile("s_wait_asynccnt 0" ::: "memory")
#endif

#if defined(__AMDGCN__) && __has_builtin(__builtin_amdgcn_s_wait_tensorcnt)
#define WAIT_TENSOR0() __builtin_amdgcn_s_wait_tensorcnt(0)
#else
#define WAIT_TENSOR0() asm volatile("s_wait_tensorcnt 0" ::: "memory")
#endif

// ---- prologue: build the two sparse IIR taps ------------------------------
// params layout (f32/i32 in d_ws): [0]=a1  [1]=a2  [2]=d1(int)  [3]=d2(int)
__global__ void ks_coeff_kernel(const float* __restrict__ gumbel,
                                const float* __restrict__ dparam,
                                const float* __restrict__ fgain,
                                const float* __restrict__ refl,
                                float* __restrict__ params) {
  if (threadIdx.x != 0 || blockIdx.x != 0) return;
  int best = 0;
  float bv = dparam[0] + gumbel[0];
  for (int j = 1; j < NDELAY; ++j) {
    float v = dparam[j] + gumbel[j];
    if (v > bv) { bv = v; best = j; }
  }
  // TAU == 0  =>  _resonant_activation(x, 0) == tanh(x); input is tanh(rc)
  float k1 = tanhf(tanhf(refl[0]));
  float k2 = tanhf(tanhf(refl[1]));
  float a1 = k1 * (1.0f - k2);
  float a2 = fminf(fmaxf(k2, -0.999f), 0.999f);
  float bnd = 0.999f - fabsf(a2);
  a1 = fminf(fmaxf(a1, -bnd), bnd);
  float sg = 1.0f / (1.0f + expf(-fgain[0]));
  float g = powf(sg, 0.45f);
  a1 *= g;
  a2 *= g;
  params[0] = a1;
  params[1] = a2;
  int* ip = (int*)params;
  ip[2] = 61 + best;                   // delay of a1 tap
  ip[3] = 61 + ((best + 1) % NDELAY);  // delay of a2 tap (handles wrap)
}

// ---- main: chunked-parallel sparse IIR, one workgroup per row -------------
// y[t] = x[t] - a1*y[t-d1] - a2*y[t-d2];  chunks of C=min(d1,d2)>=61 are
// fully parallel given earlier outputs, so in-place LDS update is safe with
// one barrier per chunk. Row load: TDM tensor_load_to_lds (one DMA per block);
// row store: async store-from-LDS b128.
__global__ __launch_bounds__(128)
void ks_lpc_kernel(const float* __restrict__ x,
                   const float* __restrict__ params,
                   float* __restrict__ y) {
  __shared__ __align__(16) float buf[T_LEN];

  const int row = blockIdx.x;
  const float a1 = params[0];
  const float a2 = params[1];
  const int d1 = ((const int*)params)[2];
  const int d2 = ((const int*)params)[3];
  const int C = d1 < d2 ? d1 : d2;

  const float* xr = x + (size_t)row * T_LEN;
  float* yr = y + (size_t)row * T_LEN;

  // ---- load whole row (32KB) into LDS via Tensor Data Mover ----
#if defined(__gfx1250__)
  if (threadIdx.x == 0) {
    // D# group 0 (ISA 8.3): count=1 | lds_addr | global_addr[56:0] | type=2
    unsigned lds_base = (unsigned)(unsigned long long)(as3_float_ptr)&buf[0];
    unsigned long long ga = (unsigned long long)xr;
    u32x4 g0;
    g0[0] = 1u;                                   // count=1, user descriptor
    g0[1] = lds_base;                             // lds_addr (bytes)
    g0[2] = (unsigned)ga;                         // global_addr[31:0]
    g0[3] = ((unsigned)(ga >> 32) & 0x01FFFFFFu)  // global_addr[56:32]
            | (2u << 30);                         // type=2 ("image")
    // D# group 1 (ISA 8.4): mask=0, data_size=4B, dims/strides for 8192x1 tile
    u32x8 g1;
    g1[0] = 2u << 16;                 // workgroup_mask=0, data_size=2 (4B)
    g1[1] = (T_LEN & 0xFFFFu) << 16;  // tensor_dim0[15:0] in [63:48]
    g1[2] = (T_LEN >> 16) | (1u << 16); // tensor_dim0[31:16] | tensor_dim1[15:0]=1
    g1[3] = ((unsigned)T_LEN) << 16;  // tensor_dim1[31:16]=0 | tile_dim0=8192
    g1[4] = 1u;                       // tile_dim1=1, tile_dim2=0
    g1[5] = (unsigned)T_LEN;          // tensor_dim0_stride[31:0]
    g1[6] = 0u;                       // stride[47:32]=0, tensor_dim1_stride lo=0
    g1[7] = 0u;                       // tensor_dim1_stride hi=0
    asm volatile("tensor_load_to_lds %0, %1" :: "s"(g0), "s"(g1) : "memory");
    WAIT_TENSOR0();                   // TENSORcnt -> 0: DMA landed in LDS
  }
#else
  for (int i = threadIdx.x; i < T_LEN / 4; i += blockDim.x) {
    reinterpret_cast<float4*>(buf)[i] = reinterpret_cast<const float4*>(xr)[i];
  }
#endif
  __syncthreads();   // publish LDS row to all 4 waves

  // ---- sequential chunks, parallel within chunk ----
  int pos = 0;
  while (pos < T_LEN) {
    int end = pos + C;
    if (end > T_LEN) end = T_LEN;
    for (int t = pos + (int)threadIdx.x; t < end; t += (int)blockDim.x) {
      float v = buf[t];
      if (t >= d1) v -= a1 * buf[t - d1];
      if (t >= d2) v -= a2 * buf[t - d2];
      buf[t] = v;  // writes in [pos,end) never read within this chunk
    }
    __syncthreads();
    pos = end;
  }

  // ---- store row back via async LDS->global ----
#if HAVE_ASYNC_LDS
  for (int i = threadIdx.x; i < T_LEN / 4; i += blockDim.x) {
    __builtin_amdgcn_global_store_async_from_lds_b128(
        (as1_v4i_ptr)(yr + i * 4), (as3_v4i_ptr)(&buf[i * 4]),
        /*offset=*/0, /*cpol=*/0);
  }
  WAIT_ASYNC0();
#else
  for (int i = threadIdx.x; i < T_LEN / 4; i += blockDim.x) {
    reinterpret_cast<float4*>(yr)[i] = reinterpret_cast<const float4*>(buf)[i];
  }
#endif
}

extern "C" void kernel_launch(void* const* d_in, const int* in_sizes, int n_in,
                              void* d_out, int out_size, void* d_ws, size_t ws_size,
                              hipStream_t stream) {
  // setup_inputs() order:
  // 0: excitation (128*1*8192 f32), 1: gumbel_noise (360 f32),
  // 2: delay_param (360 f32), 3: feedback_gain (1 f32), 4: reflection_coeffs (2 f32)
  const float* excitation = (const float*)d_in[0];
  const float* gumbel     = (const float*)d_in[1];
  const float* dparam     = (const float*)d_in[2];
  const float* fgain      = (const float*)d_in[3];
  const float* refl       = (const float*)d_in[4];
  float* out    = (float*)d_out;
  float* params = (float*)d_ws;   // 4 x 32-bit

  ks_coeff_kernel<<<1, 1, 0, stream>>>(gumbel, dparam, fgain, refl, params);
  ks_lpc_kernel<<<NROWS, 128, 0, stream>>>(excitation, params, out);
}